// DFNDLoss_22239340658777
// MI455X (gfx1250) — compile-verified
//
#include <hip/hip_runtime.h>
#include <hip/hip_bf16.h>
#include <stdint.h>

#define C_CLASSES 1000
#define C4 250   // 1000 floats = 250 float4 per row

typedef float        v2f __attribute__((ext_vector_type(2)));
typedef float        v4f __attribute__((ext_vector_type(4)));
typedef float        v8f __attribute__((ext_vector_type(8)));
typedef unsigned int v4u __attribute__((ext_vector_type(4)));
typedef int          v4i __attribute__((ext_vector_type(4)));
typedef int          v8i __attribute__((ext_vector_type(8)));

// ---------------------------------------------------------------------------
// Kernel 0: build transposed noisy matrix Mt (C x C), Mt[j][c] = M[c][j].
// ---------------------------------------------------------------------------
__global__ __launch_bounds__(256) void dfnd_build_mt(
    const float* __restrict__ na, float* __restrict__ Mt) {
    __shared__ float red[256];
    const int c = blockIdx.x;
    const int t = threadIdx.x;
    const float* row = na + (size_t)c * (C_CLASSES - 1);

    float v[4];
    float mx = -3.4e38f;
#pragma unroll
    for (int u = 0; u < 4; ++u) {
        int k = t + 256 * u;
        v[u] = (k < C_CLASSES - 1) ? row[k] : -3.4e38f;
        mx = fmaxf(mx, v[u]);
    }
    red[t] = mx;
    for (int s = 128; s > 0; s >>= 1) { __syncthreads(); if (t < s) red[t] = fmaxf(red[t], red[t + s]); }
    __syncthreads();
    const float M = red[0];
    __syncthreads();

    float e[4];
    float z = 0.f;
#pragma unroll
    for (int u = 0; u < 4; ++u) {
        int k = t + 256 * u;
        e[u] = (k < C_CLASSES - 1) ? __expf(v[u] - M) : 0.f;
        z += e[u];
    }
    red[t] = z;
    for (int s = 128; s > 0; s >>= 1) { __syncthreads(); if (t < s) red[t] += red[t + s]; }
    __syncthreads();
    const float scale = 0.05f / red[0];

#pragma unroll
    for (int u = 0; u < 4; ++u) {
        int k = t + 256 * u;
        if (k < C_CLASSES - 1) {
            int j = (k < c) ? k : k + 1;          // inverse of src_idx = j - (j>c)
            Mt[(size_t)j * C_CLASSES + c] = e[u] * scale;
        }
    }
    if (t == 0) Mt[(size_t)c * C_CLASSES + c] = 0.95f;
}

// ---------------------------------------------------------------------------
// Kernel 1: one block per sample row. NT b128 single-pass stream of T/S rows;
// TDM gathers Mt[pred,:] (4000B) into LDS overlapped with reductions.
// ---------------------------------------------------------------------------
__global__ __launch_bounds__(256) void dfnd_rows(
    const float* __restrict__ T, const float* __restrict__ S,
    const float* __restrict__ Mt,
    float* __restrict__ loss_t, float* __restrict__ klrow,
    float* __restrict__ nllrow) {
    __shared__ float sMt[C_CLASSES];   // TDM destination, 4000 B
    __shared__ float ra[256], rb[256], rc[256];
    __shared__ int   ri[256];

    const int  i   = blockIdx.x;
    const int  t   = threadIdx.x;
    const bool act = (t < C4);

    const v4f* T4 = (const v4f*)(T + (size_t)i * C_CLASSES);
    const v4f* S4 = (const v4f*)(S + (size_t)i * C_CLASSES);
    v4f tv = {0.f, 0.f, 0.f, 0.f};
    v4f sv = {0.f, 0.f, 0.f, 0.f};
    if (act) {
        tv = __builtin_nontemporal_load(T4 + t);   // TH=NT: read-once stream
        sv = __builtin_nontemporal_load(S4 + t);
    }

    // --- max/argmax(T), max(S) ---
    float mT = -3.4e38f, mS = -3.4e38f;
    int   aT = 0;
    if (act) {
#pragma unroll
        for (int j = 0; j < 4; ++j) {
            float x = tv[j];
            if (x > mT) { mT = x; aT = 4 * t + j; }
            mS = fmaxf(mS, sv[j]);
        }
    }
    ra[t] = mT; ri[t] = aT; rb[t] = mS;
    for (int s = 128; s > 0; s >>= 1) {
        __syncthreads();
        if (t < s) {
            float o = ra[t + s]; int oi = ri[t + s];
            if (o > ra[t] || (o == ra[t] && oi < ri[t])) { ra[t] = o; ri[t] = oi; }
            rb[t] = fmaxf(rb[t], rb[t + s]);
        }
    }
    __syncthreads();
    const float maxT = ra[0];
    const int   pred = ri[0];
    const float maxS = rb[0];
    __syncthreads();

    // --- wave 0 issues TDM gather of Mt[pred, 0:1000] into LDS ---
    if (t < 32) {
        unsigned lds_off = (unsigned)(uintptr_t)(&sMt[0]);  // low 32b = LDS offset
        unsigned long long ga =
            (unsigned long long)(uintptr_t)(Mt + (size_t)pred * C_CLASSES);
        // D# group0: count=1 | lds_addr | global_addr[56:0] | type=2 @ [127:126]
        unsigned g0x = (unsigned)__builtin_amdgcn_readfirstlane((int)1u);
        unsigned g0y = (unsigned)__builtin_amdgcn_readfirstlane((int)lds_off);
        unsigned g0z = (unsigned)__builtin_amdgcn_readfirstlane((int)(unsigned)ga);
        unsigned g0w = (unsigned)__builtin_amdgcn_readfirstlane(
            (int)(((unsigned)(ga >> 32) & 0x01FFFFFFu) | 0x80000000u));
        v4u g0 = {g0x, g0y, g0z, g0w};
        // D# group1: data_size=4B; tensor_dim0=1000; tile_dim0=1000;
        // tile_dim1/2 = 0 (1D); tensor_dim0_stride = 1000
        v8i g1 = {(int)(2u << 16),
                  (int)(1000u << 16),
                  0,
                  (int)(1000u << 16),
                  0,
                  1000,
                  0,
                  0};
        v4i z4 = {0, 0, 0, 0};
        v8i z8 = {0, 0, 0, 0, 0, 0, 0, 0};
        __builtin_amdgcn_tensor_load_to_lds(g0, g1, z4, z4, z8, 0);
    }

    // --- overlap: ZT, CT, ZS partial sums while TDM runs ---
    float eS[4] = {0.f, 0.f, 0.f, 0.f};
    float zT = 0.f, ct = 0.f, zS = 0.f;
    if (act) {
#pragma unroll
        for (int j = 0; j < 4; ++j) {
            float eT = __expf(tv[j] - maxT);
            eS[j]    = __expf(sv[j] - maxS);
            zT += eT;
            ct += eT * (tv[j] - sv[j]);
            zS += eS[j];
        }
    }
    ra[t] = zT; rb[t] = ct; rc[t] = zS;
    for (int s = 128; s > 0; s >>= 1) {
        __syncthreads();
        if (t < s) { ra[t] += ra[t + s]; rb[t] += rb[t + s]; rc[t] += rc[t + s]; }
    }
    __syncthreads();
    const float ZT = ra[0], CT = rb[0], ZS = rc[0];

    if (t < 32) __builtin_amdgcn_s_wait_tensorcnt(0);   // TDM done -> LDS valid
    __syncthreads();

    // --- dot(softmax_S numerators, Mt[pred,:]) ---
    float dp = 0.f;
    if (act) {
#pragma unroll
        for (int j = 0; j < 4; ++j) dp += eS[j] * sMt[4 * t + j];
    }
    ra[t] = dp;
    for (int s = 128; s > 0; s >>= 1) { __syncthreads(); if (t < s) ra[t] += ra[t + s]; }
    __syncthreads();

    if (t == 0) {
        const float dot = ra[0];
        const float lZT = __logf(ZT), lZS = __logf(ZS);
        loss_t[i] = lZT;                                     // logsumexp - max
        klrow[i]  = CT / ZT + (maxS - maxT) + (lZS - lZT);   // row KL
        nllrow[i] = lZS - __logf(dot);                       // -log adapt[i,pred]
    }
}

// ---------------------------------------------------------------------------
// Kernel 2: exact k-th smallest threshold (binary search on float bits),
// then WMMA-based deterministic reductions: one wave feeds 16x4 f32 A-tiles
// of (masked klrow | nllrow) against an all-ones 4x16 B into
// V_WMMA_F32_16X16X4_F32; column 0 of D holds the 16 row-partials.
// ---------------------------------------------------------------------------
__device__ __forceinline__ float sel_kl_val(
    const float* __restrict__ loss_t, const float* __restrict__ klrow,
    int i, unsigned thr, unsigned need) {
    unsigned u = __float_as_uint(loss_t[i]);
    bool sel = (u < thr);
    if (u == thr) {                        // stable tie-break by index
        unsigned r = 0;
        for (int j = 0; j < i; ++j)
            r += (__float_as_uint(loss_t[j]) == thr) ? 1u : 0u;
        sel = (r < need);
    }
    return sel ? klrow[i] : 0.f;
}

__global__ __launch_bounds__(1024) void dfnd_final(
    const float* __restrict__ loss_t, const float* __restrict__ klrow,
    const float* __restrict__ nllrow, float* __restrict__ out,
    int N, int K) {
    __shared__ unsigned ur[1024];
    __shared__ float    res[4];
    const int t = threadIdx.x;

    // --- phase 1: k-th smallest loss_t key (loss_t >= 0 -> bits ordered) ---
    unsigned lo = 0u, hi = 0xFFFFFFFFu;
    for (int it = 0; it < 32; ++it) {
        unsigned mid = lo + ((hi - lo) >> 1);
        unsigned c = 0;
        for (int i = t; i < N; i += 1024)
            c += (__float_as_uint(loss_t[i]) <= mid) ? 1u : 0u;
        ur[t] = c;
        for (int s = 512; s > 0; s >>= 1) { __syncthreads(); if (t < s) ur[t] += ur[t + s]; }
        __syncthreads();
        unsigned cnt = ur[0];
        __syncthreads();
        if (cnt >= (unsigned)K) hi = mid; else lo = mid + 1u;
    }
    const unsigned thr = lo;

    unsigned c = 0;
    for (int i = t; i < N; i += 1024)
        c += (__float_as_uint(loss_t[i]) < thr) ? 1u : 0u;
    ur[t] = c;
    for (int s = 512; s > 0; s >>= 1) { __syncthreads(); if (t < s) ur[t] += ur[t + s]; }
    __syncthreads();
    const unsigned need = (unsigned)K - ur[0];
    __syncthreads();

    // --- phase 2: wave 0 WMMA reductions (N is a multiple of 64) ---
    if (t < 32) {
        v8f accK = {0.f, 0.f, 0.f, 0.f, 0.f, 0.f, 0.f, 0.f};
        v8f accN = {0.f, 0.f, 0.f, 0.f, 0.f, 0.f, 0.f, 0.f};
        const v2f bones = {1.f, 1.f};
        // A 16x4 f32 layout: lanes 0-15 hold (M=t, K=0/1), lanes 16-31 (M=t-16, K=2/3)
        const int m  = (t < 16) ? t : (t - 16);
        const int k0 = (t < 16) ? 0 : 2;
        for (int base = 0; base < N; base += 64) {
            const int i0 = base + m * 4 + k0;
            const int i1 = i0 + 1;
            v2f aK = {sel_kl_val(loss_t, klrow, i0, thr, need),
                      sel_kl_val(loss_t, klrow, i1, thr, need)};
            v2f aN = {nllrow[i0], nllrow[i1]};
            accK = __builtin_amdgcn_wmma_f32_16x16x4_f32(
                false, aK, false, bones, (short)0, accK, false, false);
            accN = __builtin_amdgcn_wmma_f32_16x16x4_f32(
                false, aN, false, bones, (short)0, accN, false, false);
        }
        // D col 0: lane 0 holds M=0..7 (vgpr 0..7), lane 16 holds M=8..15
        if (t == 0) {
            float sK = 0.f, sN = 0.f;
#pragma unroll
            for (int r = 0; r < 8; ++r) { sK += accK[r]; sN += accN[r]; }
            res[0] = sK; res[1] = sN;
        }
        if (t == 16) {
            float sK = 0.f, sN = 0.f;
#pragma unroll
            for (int r = 0; r < 8; ++r) { sK += accK[r]; sN += accN[r]; }
            res[2] = sK; res[3] = sN;
        }
    }
    __syncthreads();
    if (t == 0)
        out[0] = (res[0] + res[1] + res[2] + res[3]) / (float)N;  // TAU^2=W=1
}

// ---------------------------------------------------------------------------
extern "C" void kernel_launch(void* const* d_in, const int* in_sizes, int n_in,
                              void* d_out, int out_size, void* d_ws, size_t ws_size,
                              hipStream_t stream) {
    const float* preds_S = (const float*)d_in[0];
    const float* preds_T = (const float*)d_in[1];
    const float* na      = (const float*)d_in[2];
    float* out = (float*)d_out;

    const int N = in_sizes[0] / C_CLASSES;   // 16384
    const int K = N / 2;                     // BATCH_SELECT = 0.5

    float* Mt     = (float*)d_ws;                                // 1000*1000 f32 (4 MB)
    float* lossT  = Mt + (size_t)C_CLASSES * C_CLASSES;          // N
    float* klrow  = lossT + N;                                   // N
    float* nllrow = klrow + N;                                   // N

    dfnd_build_mt<<<C_CLASSES, 256, 0, stream>>>(na, Mt);
    dfnd_rows<<<N, 256, 0, stream>>>(preds_T, preds_S, Mt, lossT, klrow, nllrow);
    dfnd_final<<<1, 1024, 0, stream>>>(lossT, klrow, nllrow, out, N, K);
}